// DynamicTemporalHeteroGNN_22033182228529
// MI455X (gfx1250) — compile-verified
//
#include <hip/hip_runtime.h>
#include <math.h>

#define H 32
#define NS 100000
#define NQ 50000
#define NC 5000
#define E_ATT 2000000
#define E_TAG 150000
#define E_PRE 100000
#define E_MAST 500000
#define E_LBL 500000

typedef __attribute__((ext_vector_type(2))) float v2f;
typedef __attribute__((ext_vector_type(8))) float v8f;

static inline int cdiv(int a, int b) { return (a + b - 1) / b; }

// ---------------- order-preserving float<->uint encoding (for atomic max) ----
__device__ __forceinline__ unsigned fenc(float f) {
  unsigned u = __float_as_uint(f);
  return (u & 0x80000000u) ? ~u : (u | 0x80000000u);
}
__device__ __forceinline__ float fdec(unsigned u) {
  unsigned b = (u & 0x80000000u) ? (u ^ 0x80000000u) : ~u;
  return __uint_as_float(b);
}
__device__ __forceinline__ float sigmoidf_(float x) { return 1.f / (1.f + __expf(-x)); }

// ---------------- generic fill ----------------------------------------------
__global__ void fill_kernel(float* __restrict__ p, float v, int n) {
  int i = blockIdx.x * blockDim.x + threadIdx.x;
  if (i < n) p[i] = v;
}

// ---------------- encoders: y[n,H] = x[n,ind] @ w[ind,H] + b ----------------
__global__ void encode_kernel(const float* __restrict__ x, int ind,
                              const float* __restrict__ w, const float* __restrict__ b,
                              float* __restrict__ y, int n) {
  int idx = blockIdx.x * blockDim.x + threadIdx.x;
  if (idx >= n * H) return;
  int node = idx >> 5, j = idx & 31;
  float s = b[j];
  for (int i = 0; i < ind; ++i) s += x[node * ind + i] * w[i * H + j];
  y[idx] = s;
}

// ---------------- GRUCell (gate order r,z,n) --------------------------------
__global__ void gru_kernel(const float* __restrict__ xin, const float* __restrict__ h,
                           const float* __restrict__ wih, const float* __restrict__ whh,
                           const float* __restrict__ bih, const float* __restrict__ bhh,
                           float* __restrict__ y, int n) {
  int idx = blockIdx.x * blockDim.x + threadIdx.x;
  if (idx >= n * H) return;
  int node = idx >> 5, j = idx & 31;
  float gir = bih[j], giz = bih[H + j], gin = bih[2 * H + j];
  float ghr = bhh[j], ghz = bhh[H + j], ghn = bhh[2 * H + j];
  for (int i = 0; i < H; ++i) {
    float xv = xin[node * H + i], hv = h[node * H + i];
    gir += xv * wih[i * 3 * H + j];
    giz += xv * wih[i * 3 * H + H + j];
    gin += xv * wih[i * 3 * H + 2 * H + j];
    ghr += hv * whh[i * 3 * H + j];
    ghz += hv * whh[i * 3 * H + H + j];
    ghn += hv * whh[i * 3 * H + 2 * H + j];
  }
  float r = sigmoidf_(gir + ghr);
  float z = sigmoidf_(giz + ghz);
  float nn = tanhf(gin + r * ghn);
  y[idx] = (1.f - z) * nn + z * h[idx];
}

// ---------------- WMMA f32: Y[n,32] = X[n,32] @ W[32,32] --------------------
// Grid-stride over 16-row tiles. Each wave32 hoists the full B fragment set
// (8 K-steps x 2 N-tiles, v2f each = 32 VGPRs) into registers once, then per
// tile issues 8 contiguous global_load_b64 A-loads and 16 V_WMMA_F32_16X16X4_F32.
// A rows are clamped (not zeroed/predicated): output row M depends only on A
// row M. Stores are UNGUARDED: Y must have capacity for ceil(n/16)*16 rows
// (always true here: Y is the NS-row scratch buffer). No EXEC divergence at all.
__global__ __launch_bounds__(128) void wmma_xform_kernel(
    const float* __restrict__ X, const float* __restrict__ W,
    float* __restrict__ Y, int nrows) {
  const int lane = threadIdx.x & 31;
  const int half = lane >> 4;
  const int lo = lane & 15;
  const int nTiles = (nrows + 15) >> 4;
  const int wavesTotal = gridDim.x * 4;
  const int waveId = blockIdx.x * 4 + (threadIdx.x >> 5);

  // Hoist B fragments: bf[k][t] covers K rows {k*4+2*half, +1}, cols t*16+lo.
  v2f bf[8][2];
#pragma unroll
  for (int ks = 0; ks < 8; ++ks) {
    const int kr = ks * 4 + 2 * half;
#pragma unroll
    for (int t = 0; t < 2; ++t) {
      bf[ks][t].x = W[kr * H + t * 16 + lo];
      bf[ks][t].y = W[(kr + 1) * H + t * 16 + lo];
    }
  }

  for (int tile = waveId; tile < nTiles; tile += wavesTotal) {
    const int rowBase = tile * 16;
    int row = rowBase + lo;
    row = (row < nrows) ? row : (nrows - 1);  // clamp: no divergence needed
    v8f c0 = {};
    v8f c1 = {};
#pragma unroll
    for (int ks = 0; ks < 8; ++ks) {
      const int kr = ks * 4 + 2 * half;
      const v2f a = *(const v2f*)(X + (size_t)row * H + kr);  // 8B aligned (kr even)
      c0 = __builtin_amdgcn_wmma_f32_16x16x4_f32(false, a, false, bf[ks][0], (short)0, c0,
                                                 false, false);
      c1 = __builtin_amdgcn_wmma_f32_16x16x4_f32(false, a, false, bf[ks][1], (short)0, c1,
                                                 false, false);
    }
    // Unguarded stores: one base address, 16 immediate-offset b32 stores.
    float* __restrict__ yb = Y + (size_t)(rowBase + 8 * half) * H + lo;
#pragma unroll
    for (int v = 0; v < 8; ++v) {
      yb[v * H] = c0[v];
      yb[v * H + 16] = c1[v];
    }
  }
}

// ---------------- fold: fold[i] = sum_j W[i][j]*att[j]  (src then dst) ------
__global__ void fold_kernel(const float* __restrict__ wS, const float* __restrict__ aS,
                            const float* __restrict__ wD, const float* __restrict__ aD,
                            float* __restrict__ fold) {
  int t = threadIdx.x;  // 64 threads
  const float* w = (t < H) ? wS : wD;
  const float* a = (t < H) ? aS : aD;
  int i = t & 31;
  float s = 0.f;
  for (int j = 0; j < H; ++j) s += w[i * H + j] * a[j];
  fold[t] = s;
}

// ---------------- per-node attention scalar: a[n] = x[n,:] . fold -----------
__global__ void node_score_kernel(const float* __restrict__ x, const float* __restrict__ fold,
                                  float* __restrict__ out, int n) {
  int i = blockIdx.x * blockDim.x + threadIdx.x;
  if (i >= n) return;
  float s = 0.f;
  for (int j = 0; j < H; ++j) s += x[i * H + j] * fold[j];
  out[i] = s;
}

// ---------------- pass A: segment max of leaky_relu scores ------------------
__global__ void edge_max_kernel(const int* __restrict__ src, const int* __restrict__ dst,
                                const float* __restrict__ aS, const float* __restrict__ aD,
                                unsigned* __restrict__ mEnc, int E) {
  int e = blockIdx.x * blockDim.x + threadIdx.x;
  if (e >= E) return;
  float v = aS[src[e]] + aD[dst[e]];
  v = (v >= 0.f) ? v : 0.2f * v;
  atomicMax(&mEnc[dst[e]], fenc(v));
}

// ---------------- pass B: ex = exp(e-m); denom += ex; acc += ex*hs[src] -----
__global__ void edge_acc_kernel(const int* __restrict__ src, const int* __restrict__ dst,
                                const float* __restrict__ aS, const float* __restrict__ aD,
                                const unsigned* __restrict__ mEnc, const float* __restrict__ hs,
                                float* __restrict__ denom, float* __restrict__ acc, int E) {
  int idx = blockIdx.x * blockDim.x + threadIdx.x;
  if (idx >= E * H) return;
  int e = idx >> 5, k = idx & 31;
  int s = src[e], d = dst[e];
  float v = aS[s] + aD[d];
  v = (v >= 0.f) ? v : 0.2f * v;
  float ex = __expf(v - fdec(mEnc[d]));
  if (k == 0) atomicAdd(&denom[d], ex);
  atomicAdd(&acc[d * H + k], ex * hs[s * H + k]);
}

// ---------------- pass C: out += acc/denom + bias ---------------------------
__global__ void seg_finish_kernel(const float* __restrict__ acc, const float* __restrict__ denom,
                                  const float* __restrict__ bias, float* __restrict__ out, int n) {
  int idx = blockIdx.x * blockDim.x + threadIdx.x;
  if (idx >= n * H) return;
  int node = idx >> 5, k = idx & 31;
  float d = denom[node];
  float v = (d > 0.f) ? acc[idx] / d : 0.f;
  out[idx] += v + bias[k];
}

__global__ void relu_kernel(float* __restrict__ x, int n) {
  int i = blockIdx.x * blockDim.x + threadIdx.x;
  if (i < n) x[i] = fmaxf(x[i], 0.f);
}

// ---------------- corr head: warp per label edge ----------------------------
__global__ void corr_kernel(const float* __restrict__ xs, const float* __restrict__ xq,
                            const int* __restrict__ lsrc, const int* __restrict__ ldst,
                            const float* __restrict__ w1, const float* __restrict__ b1,
                            const float* __restrict__ w2, const float* __restrict__ b2,
                            float* __restrict__ logits, int E) {
  int wid = (blockIdx.x * blockDim.x + threadIdx.x) >> 5;
  int lane = threadIdx.x & 31;
  if (wid >= E) return;
  int s = lsrc[wid], q = ldst[wid];
  float h = b1[lane];
  for (int i = 0; i < H; ++i) h += xs[s * H + i] * w1[i * H + lane];
  for (int i = 0; i < H; ++i) h += xq[q * H + i] * w1[(H + i) * H + lane];
  h = fmaxf(h, 0.f);
  float p = h * w2[lane];
  for (int off = 16; off; off >>= 1) p += __shfl_xor(p, off, 32);
  if (lane == 0) logits[wid] = p + b2[0];
}

// ---------------- mastery head: warp per concept ----------------------------
__global__ void mastery_kernel(const float* __restrict__ xc,
                               const float* __restrict__ w1, const float* __restrict__ b1,
                               const float* __restrict__ w2, const float* __restrict__ b2,
                               float* __restrict__ out, int n) {
  int wid = (blockIdx.x * blockDim.x + threadIdx.x) >> 5;
  int lane = threadIdx.x & 31;
  if (wid >= n) return;
  float h = b1[lane];
  for (int i = 0; i < H; ++i) h += xc[wid * H + i] * w1[i * H + lane];
  h = fmaxf(h, 0.f);
  float p = h * w2[lane];
  for (int off = 16; off; off >>= 1) p += __shfl_xor(p, off, 32);
  if (lane == 0) out[wid] = sigmoidf_(p + b2[0]);
}

// ---------------- host-side driver ------------------------------------------
struct EdgeSpec { int pe; int st; int dt; int si; int di; int E; };
// conv leaf order (sorted keys): att=0, mast=1, prereq=2, rev_att=3, rev_mast=4, rev_tag=5, tag=6
static const EdgeSpec kEdges[7] = {
  {0, 0, 1, 4, 5,  E_ATT },   // att:      student -> question
  {3, 1, 0, 5, 4,  E_ATT },   // rev_att:  question -> student
  {6, 1, 2, 6, 7,  E_TAG },   // tag:      question -> concept
  {5, 2, 1, 7, 6,  E_TAG },   // rev_tag:  concept -> question
  {2, 2, 2, 8, 9,  E_PRE },   // prereq:   concept -> concept
  {1, 0, 2, 10, 11, E_MAST }, // mast:     student -> concept
  {4, 2, 0, 11, 10, E_MAST }, // rev_mast: concept -> student
};

extern "C" void kernel_launch(void* const* d_in, const int* in_sizes, int n_in,
                              void* d_out, int out_size, void* d_ws, size_t ws_size,
                              hipStream_t stream) {
  (void)in_sizes; (void)n_in; (void)out_size; (void)ws_size;
  const int TB = 256;

  const float* student_x = (const float*)d_in[0];
  const float* question_x = (const float*)d_in[1];
  const float* concept_x = (const float*)d_in[2];
  const float* prev_h = (const float*)d_in[3];
  auto P = [&](int i) { return (const float*)d_in[i]; };
  // params flattened at d_in[14..101] in sorted-key (jax tree) order:
  // conv1(14..48), conv2(49..83), corr_b1=84, corr_b2=85, corr_w1=86, corr_w2=87,
  // enc_concept_b=88, enc_concept_w=89, enc_question_b=90, enc_question_w=91,
  // enc_student_b=92, enc_student_w=93, gru_b_hh=94, gru_b_ih=95, gru_w_hh=96,
  // gru_w_ih=97, mast_b1=98, mast_b2=99, mast_w1=100, mast_w2=101

  // workspace layout (floats)
  float* W = (float*)d_ws;
  size_t o = 0;
  float* xs = W + o; o += (size_t)NS * H;
  float* xq = W + o; o += (size_t)NQ * H;
  float* xc = W + o; o += (size_t)NC * H;
  float* ys = W + o; o += (size_t)NS * H;
  float* yq = W + o; o += (size_t)NQ * H;
  float* yc = W + o; o += (size_t)NC * H;
  float* hsb = W + o; o += (size_t)NS * H;   // WMMA dest: capacity >= padded rows
  float* aSrc = W + o; o += (size_t)NS;
  float* aDst = W + o; o += (size_t)NS;
  float* mBuf = W + o; o += (size_t)NS;      // reinterpreted as unsigned
  float* denom = W + o; o += (size_t)NS;
  float* acc = W + o; o += (size_t)NS * H;
  float* fold = W + o; o += 64;
  unsigned* mEnc = (unsigned*)mBuf;

  // ---- encoders + GRU (hsb used as pre-GRU temp) ----
  encode_kernel<<<cdiv(NS * H, TB), TB, 0, stream>>>(student_x, 4, P(93), P(92), hsb, NS);
  gru_kernel<<<cdiv(NS * H, TB), TB, 0, stream>>>(hsb, prev_h, P(97), P(96), P(95), P(94), xs, NS);
  encode_kernel<<<cdiv(NQ * H, TB), TB, 0, stream>>>(question_x, 3, P(91), P(90), xq, NQ);
  encode_kernel<<<cdiv(NC * H, TB), TB, 0, stream>>>(concept_x, 2, P(89), P(88), xc, NC);

  float* outS = (float*)d_out;
  float* outQ = outS + (size_t)NS * H;
  float* outC = outQ + (size_t)NQ * H;
  float* logits = outC + (size_t)NC * H;
  float* mastery = logits + (size_t)E_LBL;

  const int ncnt[3] = {NS, NQ, NC};
  for (int layer = 0; layer < 2; ++layer) {
    const float* xl[3];
    float* yl[3];
    if (layer == 0) { xl[0] = xs; xl[1] = xq; xl[2] = xc; yl[0] = ys; yl[1] = yq; yl[2] = yc; }
    else            { xl[0] = ys; xl[1] = yq; xl[2] = yc; yl[0] = outS; yl[1] = outQ; yl[2] = outC; }
    const int base = (layer == 0) ? 14 : 49;

    for (int t = 0; t < 3; ++t)
      fill_kernel<<<cdiv(ncnt[t] * H, TB), TB, 0, stream>>>(yl[t], 0.f, ncnt[t] * H);

    for (int ei = 0; ei < 7; ++ei) {
      const EdgeSpec& e = kEdges[ei];
      const float* attD = P(base + e.pe * 5 + 0);
      const float* attS = P(base + e.pe * 5 + 1);
      const float* bias = P(base + e.pe * 5 + 2);
      const float* wD = P(base + e.pe * 5 + 3);
      const float* wS = P(base + e.pe * 5 + 4);
      const int nsrc = ncnt[e.st], ndst = ncnt[e.dt];
      const float* xsrc = xl[e.st];
      const float* xdst = xl[e.dt];
      const int* src = (const int*)d_in[e.si];
      const int* dst = (const int*)d_in[e.di];

      fold_kernel<<<1, 64, 0, stream>>>(wS, attS, wD, attD, fold);
      // grid-stride WMMA: cap grid so each wave reuses its hoisted B registers
      int wblocks = cdiv(nsrc, 64);
      if (wblocks > 512) wblocks = 512;
      wmma_xform_kernel<<<wblocks, 128, 0, stream>>>(xsrc, wS, hsb, nsrc);
      node_score_kernel<<<cdiv(nsrc, TB), TB, 0, stream>>>(xsrc, fold, aSrc, nsrc);
      node_score_kernel<<<cdiv(ndst, TB), TB, 0, stream>>>(xdst, fold + H, aDst, ndst);
      fill_kernel<<<cdiv(ndst, TB), TB, 0, stream>>>(mBuf, 0.f, ndst);     // uint 0
      fill_kernel<<<cdiv(ndst, TB), TB, 0, stream>>>(denom, 0.f, ndst);
      fill_kernel<<<cdiv(ndst * H, TB), TB, 0, stream>>>(acc, 0.f, ndst * H);
      edge_max_kernel<<<cdiv(e.E, TB), TB, 0, stream>>>(src, dst, aSrc, aDst, mEnc, e.E);
      edge_acc_kernel<<<cdiv(e.E * H, TB), TB, 0, stream>>>(src, dst, aSrc, aDst, mEnc, hsb,
                                                            denom, acc, e.E);
      seg_finish_kernel<<<cdiv(ndst * H, TB), TB, 0, stream>>>(acc, denom, bias, yl[e.dt], ndst);
    }
    for (int t = 0; t < 3; ++t)
      relu_kernel<<<cdiv(ncnt[t] * H, TB), TB, 0, stream>>>(yl[t], ncnt[t] * H);
  }

  // ---- heads ----
  const int* lsrc = (const int*)d_in[12];
  const int* ldst = (const int*)d_in[13];
  corr_kernel<<<cdiv(E_LBL * 32, TB), TB, 0, stream>>>(outS, outQ, lsrc, ldst,
                                                       P(86), P(84), P(87), P(85), logits, E_LBL);
  mastery_kernel<<<cdiv(NC * 32, TB), TB, 0, stream>>>(outC, P(100), P(98), P(101), P(99),
                                                       mastery, NC);
}